// DCB_75557064671368
// MI455X (gfx1250) — compile-verified
//
#include <hip/hip_runtime.h>
#include <math.h>

// ---------------- problem constants ----------------
#define HH   192
#define WW_  192
#define HWSZ (HH * WW_)          // 36864
#define NB   2
#define NPIX (NB * HWSZ)         // 73728
#define CCH  64
#define KTAP 9
#define EPSB 1e-5f

typedef __attribute__((ext_vector_type(2))) float v2f;
typedef __attribute__((ext_vector_type(8))) float v8f;

// f32 WMMA 16x16x4 fragment mapping (per CDNA5 ISA 7.12.2):
//   A (16x4, MxK): lane L -> M = L&15 ; VGPR v -> K = v + 2*(L>=16)
//   B (4x16, KxN): lane L -> N = L&15 ; VGPR v -> K = v + 2*(L>=16)
//   C/D (16x16)  : lane L -> N = L&15 ; VGPR v -> M = v + 8*(L>=16)

// ---------------- conv1x1 #1: NCHW x -> NHWC y1, via f32 WMMA ----------------
__global__ __launch_bounds__(128)
void conv1_k(const float* __restrict__ x, const float* __restrict__ w1,
             const float* __restrict__ b1, float* __restrict__ y1) {
    __shared__ float w1S[64 * 65];   // [o][c], pitch 65 (bank-conflict free A reads)
    __shared__ float Bm[64 * 16];    // [c][pos], pitch 16
    const int tid = threadIdx.x;
    const int pixbase = blockIdx.x * 16;        // 16 consecutive w positions
    const int b   = pixbase / HWSZ;
    const int rem = pixbase % HWSZ;
    const int h   = rem / WW_;
    const int w0  = rem % WW_;

    for (int u = tid; u < 4096; u += 128) {
        int o = u >> 6, c = u & 63;
        w1S[o * 65 + c] = w1[u];
    }
    {   // x[b][c][h][w0..w0+15] -> Bm[c][0..15] (coalesced float4 loads)
        int c = tid >> 1, half = tid & 1;
        const float4* src =
            (const float4*)(x + (size_t)(b * 64 + c) * HWSZ + h * WW_ + w0 + half * 8);
        float4 a = src[0], q = src[1];
        float* dst = &Bm[c * 16 + half * 8];
        dst[0]=a.x; dst[1]=a.y; dst[2]=a.z; dst[3]=a.w;
        dst[4]=q.x; dst[5]=q.y; dst[6]=q.z; dst[7]=q.w;
    }
    __syncthreads();

    const int wave = tid >> 5, lane = tid & 31;
    const int lm = lane & 15, hi = lane >> 4;
    v8f acc = {0.f,0.f,0.f,0.f,0.f,0.f,0.f,0.f};
    #pragma unroll
    for (int ki = 0; ki < 16; ++ki) {
        const int k0 = ki * 4 + 2 * hi;
        v2f af, bf;
        af[0] = w1S[(wave * 16 + lm) * 65 + k0];
        af[1] = w1S[(wave * 16 + lm) * 65 + k0 + 1];
        bf[0] = Bm[(k0    ) * 16 + lm];
        bf[1] = Bm[(k0 + 1) * 16 + lm];
        acc = __builtin_amdgcn_wmma_f32_16x16x4_f32(false, af, false, bf,
                                                    (short)0, acc, false, false);
    }
    #pragma unroll
    for (int vr = 0; vr < 8; ++vr) {
        const int ch = wave * 16 + hi * 8 + vr;
        y1[(size_t)(pixbase + lm) * CCH + ch] = acc[vr] + b1[ch];
    }
}

// ---------------- per-channel sum/sumsq partials (deterministic) ----------------
__global__ __launch_bounds__(256)
void stats_k(const float* __restrict__ in, float* __restrict__ partials) {
    __shared__ float sS[256], ssS[256];
    const int tid = threadIdx.x;
    const int ch = tid & 63, grp = tid >> 6;    // 4 pixel groups
    float s = 0.f, ss = 0.f;
    for (int pix = blockIdx.x * 4 + grp; pix < NPIX; pix += gridDim.x * 4) {
        float v = in[(size_t)pix * CCH + ch];
        s += v; ss += v * v;
    }
    sS[tid] = s; ssS[tid] = ss;
    __syncthreads();
    if (tid < 64) {
        float ts  = sS[tid]  + sS[tid + 64]  + sS[tid + 128]  + sS[tid + 192];
        float tss = ssS[tid] + ssS[tid + 64] + ssS[tid + 128] + ssS[tid + 192];
        partials[blockIdx.x * 128 + tid]      = ts;
        partials[blockIdx.x * 128 + 64 + tid] = tss;
    }
}

__global__ void reduce_k(const float* __restrict__ partials,
                         float* __restrict__ stats_out, int nblk) {
    const int t = threadIdx.x;          // 0..127 : [sum x64 | sumsq x64]
    float s = 0.f;
    for (int b = 0; b < nblk; ++b) s += partials[b * 128 + t];
    stats_out[t] = s;
}

__global__ void bnprep_k(const float* __restrict__ stats,
                         const float* __restrict__ gamma,
                         const float* __restrict__ beta,
                         float* __restrict__ bnp) {
    const int c = threadIdx.x;          // 64
    const float inv_n = 1.0f / (float)NPIX;
    float mean = stats[c] * inv_n;
    float var  = stats[64 + c] * inv_n - mean * mean;
    float sc   = gamma[c] * rsqrtf(var + EPSB);
    bnp[c]      = sc;
    bnp[64 + c] = beta[c] - mean * sc;
}

// ---------------- fused BN (+optional ReLU), NHWC in place ----------------
__global__ __launch_bounds__(256)
void bnrelu_k(float* __restrict__ buf, const float* __restrict__ bnp) {
    __shared__ float scS[64], shS[64];
    const int tid = threadIdx.x;
    if (tid < 64) { scS[tid] = bnp[tid]; shS[tid] = bnp[64 + tid]; }
    __syncthreads();
    const int total4 = NPIX * CCH / 4;
    for (int i = blockIdx.x * blockDim.x + tid; i < total4;
         i += gridDim.x * blockDim.x) {
        float4* p = (float4*)buf + i;
        float4 v = *p;
        const int c0 = (i * 4) & 63;
        v.x = fmaxf(fmaf(v.x, scS[c0 + 0], shS[c0 + 0]), 0.f);
        v.y = fmaxf(fmaf(v.y, scS[c0 + 1], shS[c0 + 1]), 0.f);
        v.z = fmaxf(fmaf(v.z, scS[c0 + 2], shS[c0 + 2]), 0.f);
        v.w = fmaxf(fmaf(v.w, scS[c0 + 3], shS[c0 + 3]), 0.f);
        *p = v;
    }
}

// ---------------- offset conv3x3: a1 NHWC -> offs [pix][18] (VALU) ----------------
__global__ __launch_bounds__(128)
void offsetconv_k(const float* __restrict__ a1, const float* __restrict__ ow,
                  const float* __restrict__ ob, float* __restrict__ offs) {
    __shared__ float wS[KTAP * 64 * 18];   // [tap][c][j]
    const int tid = threadIdx.x;
    for (int u = tid; u < KTAP * 64 * 18; u += 128) {
        int tap = u / 1152, r = u % 1152, c = r / 18, j = r % 18;
        wS[u] = ow[j * 576 + c * 9 + tap];       // ow: [18][64][3][3]
    }
    __syncthreads();
    const int pix = blockIdx.x * 128 + tid;
    const int b = pix / HWSZ, r = pix % HWSZ, h = r / WW_, w = r % WW_;
    float acc[18];
    #pragma unroll
    for (int j = 0; j < 18; ++j) acc[j] = ob[j];
    for (int tap = 0; tap < 9; ++tap) {
        const int yy = h + tap / 3 - 1, xx = w + tap % 3 - 1;
        if (yy < 0 || yy >= HH || xx < 0 || xx >= WW_) continue;   // zero pad
        const float4* ap =
            (const float4*)(a1 + (size_t)(b * HWSZ + yy * WW_ + xx) * CCH);
        const float* wt = &wS[tap * 1152];
        #pragma unroll 4
        for (int c4 = 0; c4 < 16; ++c4) {
            const float4 v = ap[c4];
            const float* w0p = wt + (c4 * 4) * 18;
            #pragma unroll
            for (int j = 0; j < 18; ++j)
                acc[j] += v.x * w0p[j] + v.y * w0p[18 + j] +
                          v.z * w0p[36 + j] + v.w * w0p[54 + j];
        }
    }
    #pragma unroll
    for (int j = 0; j < 18; ++j) offs[(size_t)pix * 18 + j] = acc[j];
}

// ---------------- deformable conv3x3: bilinear im2col -> f32 WMMA GEMM ----------------
__global__ __launch_bounds__(128)
void deform_k(const float* __restrict__ a1, const float* __restrict__ offs,
              const float* __restrict__ dw, const float* __restrict__ db,
              float* __restrict__ out) {
    __shared__ float patch[576 * 16];   // [c*9+t][pos], pitch 16
    __shared__ float dwS[64 * 33];      // [o][kl], 32-wide K chunk, pitch 33
    const int tid = threadIdx.x;
    const int pixbase = blockIdx.x * 16;
    const int b   = pixbase / HWSZ;
    const int rem = pixbase % HWSZ;
    const int h   = rem / WW_;
    const int w0  = rem % WW_;
    const float* fbase = a1 + (size_t)b * HWSZ * CCH;

    // Phase 1: bilinear gather into LDS patch. 16 groups of 8 threads,
    // each group handles one (pos,tap) pair per iteration, 8 channels/thread.
    const int grp = tid >> 3, lng = tid & 7, c0 = lng * 8;
    for (int it = 0; it < 9; ++it) {
        const int pair = it * 16 + grp;       // 0..143
        const int p = pair / 9, t = pair % 9;
        const int pix = pixbase + p;
        const float offy = offs[(size_t)pix * 18 + t * 2 + 0];
        const float offx = offs[(size_t)pix * 18 + t * 2 + 1];
        const float py = (float)(h + t / 3 - 1) + offy;
        const float px = (float)(w0 + p + t % 3 - 1) + offx;
        const float y0f = floorf(py), x0f = floorf(px);
        const float fy = py - y0f, fx = px - x0f;
        const int y0 = (int)y0f, x0 = (int)x0f, y1 = y0 + 1, x1 = x0 + 1;
        const bool vy0 = (y0 >= 0) && (y0 < HH), vy1 = (y1 >= 0) && (y1 < HH);
        const bool vx0 = (x0 >= 0) && (x0 < WW_), vx1 = (x1 >= 0) && (x1 < WW_);
        const float w00 = (1.f - fy) * (1.f - fx), w01 = (1.f - fy) * fx;
        const float w10 = fy * (1.f - fx),         w11 = fy * fx;
        float v[8];
        #pragma unroll
        for (int j = 0; j < 8; ++j) v[j] = 0.f;
        auto accum = [&](int yy, int xx, float wgt, bool valid) {
            if (!valid) return;
            const float4* q =
                (const float4*)(fbase + (size_t)(yy * WW_ + xx) * CCH + c0);
            const float4 a = q[0], bb = q[1];
            v[0] += wgt * a.x;  v[1] += wgt * a.y;
            v[2] += wgt * a.z;  v[3] += wgt * a.w;
            v[4] += wgt * bb.x; v[5] += wgt * bb.y;
            v[6] += wgt * bb.z; v[7] += wgt * bb.w;
        };
        accum(y0, x0, w00, vy0 && vx0);
        accum(y0, x1, w01, vy0 && vx1);
        accum(y1, x0, w10, vy1 && vx0);
        accum(y1, x1, w11, vy1 && vx1);
        #pragma unroll
        for (int j = 0; j < 8; ++j)
            patch[((c0 + j) * 9 + t) * 16 + p] = v[j];
    }
    __syncthreads();

    // Phase 2: D(64x16) = dw(64x576) * patch(576x16), chunked K staging of dw.
    const int wave = tid >> 5, lane = tid & 31;
    const int lm = lane & 15, hi = lane >> 4;
    v8f acc = {0.f,0.f,0.f,0.f,0.f,0.f,0.f,0.f};
    for (int chunk = 0; chunk < 18; ++chunk) {
        __syncthreads();
        for (int u = tid; u < 2048; u += 128) {
            const int ch = u >> 5, kl = u & 31;
            dwS[ch * 33 + kl] = dw[(size_t)ch * 576 + chunk * 32 + kl];
        }
        __syncthreads();
        #pragma unroll
        for (int ki = 0; ki < 8; ++ki) {
            const int kl = ki * 4 + 2 * hi;
            const int kg = chunk * 32 + kl;
            v2f af, bf;
            af[0] = dwS[(wave * 16 + lm) * 33 + kl];
            af[1] = dwS[(wave * 16 + lm) * 33 + kl + 1];
            bf[0] = patch[(kg    ) * 16 + lm];
            bf[1] = patch[(kg + 1) * 16 + lm];
            acc = __builtin_amdgcn_wmma_f32_16x16x4_f32(false, af, false, bf,
                                                        (short)0, acc, false, false);
        }
    }
    #pragma unroll
    for (int vr = 0; vr < 8; ++vr) {
        const int ch = wave * 16 + hi * 8 + vr;
        out[(size_t)(pixbase + lm) * CCH + ch] = acc[vr] + db[ch];
    }
}

// ---------------- conv1x1 #2: NHWC in -> NHWC y2, via f32 WMMA ----------------
__global__ __launch_bounds__(128)
void conv2_k(const float* __restrict__ in, const float* __restrict__ w2,
             const float* __restrict__ b2, float* __restrict__ y2) {
    __shared__ float w2S[64 * 65];
    __shared__ float Bm[64 * 16];
    const int tid = threadIdx.x;
    const int pixbase = blockIdx.x * 16;
    for (int u = tid; u < 4096; u += 128) {
        int o = u >> 6, c = u & 63;
        w2S[o * 65 + c] = w2[u];
    }
    {   // NHWC tile -> Bm[c][p] (transpose via LDS)
        const int p = tid >> 3, cc = (tid & 7) * 8;
        const float4* src = (const float4*)(in + (size_t)(pixbase + p) * CCH + cc);
        const float4 a = src[0], q = src[1];
        Bm[(cc + 0) * 16 + p] = a.x; Bm[(cc + 1) * 16 + p] = a.y;
        Bm[(cc + 2) * 16 + p] = a.z; Bm[(cc + 3) * 16 + p] = a.w;
        Bm[(cc + 4) * 16 + p] = q.x; Bm[(cc + 5) * 16 + p] = q.y;
        Bm[(cc + 6) * 16 + p] = q.z; Bm[(cc + 7) * 16 + p] = q.w;
    }
    __syncthreads();
    const int wave = tid >> 5, lane = tid & 31;
    const int lm = lane & 15, hi = lane >> 4;
    v8f acc = {0.f,0.f,0.f,0.f,0.f,0.f,0.f,0.f};
    #pragma unroll
    for (int ki = 0; ki < 16; ++ki) {
        const int k0 = ki * 4 + 2 * hi;
        v2f af, bf;
        af[0] = w2S[(wave * 16 + lm) * 65 + k0];
        af[1] = w2S[(wave * 16 + lm) * 65 + k0 + 1];
        bf[0] = Bm[(k0    ) * 16 + lm];
        bf[1] = Bm[(k0 + 1) * 16 + lm];
        acc = __builtin_amdgcn_wmma_f32_16x16x4_f32(false, af, false, bf,
                                                    (short)0, acc, false, false);
    }
    #pragma unroll
    for (int vr = 0; vr < 8; ++vr) {
        const int ch = wave * 16 + hi * 8 + vr;
        y2[(size_t)(pixbase + lm) * CCH + ch] = acc[vr] + b2[ch];
    }
}

// ---------------- final BN2 + NHWC -> NCHW transpose ----------------
__global__ __launch_bounds__(256)
void final_k(const float* __restrict__ y2, const float* __restrict__ bnp,
             float* __restrict__ out) {
    __shared__ float scS[64], shS[64];
    const int tid = threadIdx.x;
    if (tid < 64) { scS[tid] = bnp[tid]; shS[tid] = bnp[64 + tid]; }
    __syncthreads();
    const int total4 = NPIX * CCH / 4;
    for (int i = blockIdx.x * blockDim.x + tid; i < total4;
         i += gridDim.x * blockDim.x) {
        const int e = i * 4;                    // NCHW flat element index
        const int b = e / (CCH * HWSZ);
        const int r = e % (CCH * HWSZ);
        const int o = r / HWSZ;
        const int hw = r % HWSZ;
        const float* src = y2 + (size_t)(b * HWSZ + hw) * CCH + o;
        const float sc = scS[o], sh = shS[o];
        float4 v;
        v.x = fmaf(src[0 * CCH], sc, sh);
        v.y = fmaf(src[1 * CCH], sc, sh);
        v.z = fmaf(src[2 * CCH], sc, sh);
        v.w = fmaf(src[3 * CCH], sc, sh);
        *((float4*)out + i) = v;
    }
}

// ---------------- host side ----------------
extern "C" void kernel_launch(void* const* d_in, const int* in_sizes, int n_in,
                              void* d_out, int out_size, void* d_ws, size_t ws_size,
                              hipStream_t stream) {
    const float* x   = (const float*)d_in[0];
    const float* w1  = (const float*)d_in[1];
    const float* b1  = (const float*)d_in[2];
    const float* g1  = (const float*)d_in[3];
    const float* be1 = (const float*)d_in[4];
    const float* ow  = (const float*)d_in[5];
    const float* ob  = (const float*)d_in[6];
    const float* dw  = (const float*)d_in[7];
    const float* db  = (const float*)d_in[8];
    const float* w2  = (const float*)d_in[9];
    const float* b2  = (const float*)d_in[10];
    const float* g2  = (const float*)d_in[11];
    const float* be2 = (const float*)d_in[12];
    float* out = (float*)d_out;

    // workspace layout (floats)
    float* ws = (float*)d_ws;
    float* buf1     = ws;                                  // NPIX*64  (y1/a1, then y2)
    float* buf2     = buf1 + (size_t)NPIX * CCH;           // NPIX*64  (deform out)
    float* offsbuf  = buf2 + (size_t)NPIX * CCH;           // NPIX*18
    float* partials = offsbuf + (size_t)NPIX * 18;         // 512*128
    float* stats    = partials + 512 * 128;                // 2 x 128
    float* bnp      = stats + 256;                         // 2 x 128

    const int NTILE = NPIX / 16;     // 4608
    const int NSTAT = 512;

    // 1) conv1x1 #1 (WMMA) -> buf1 (NHWC)
    conv1_k<<<NTILE, 128, 0, stream>>>(x, w1, b1, buf1);
    // 2) BN1 stats (deterministic two-stage)
    stats_k<<<NSTAT, 256, 0, stream>>>(buf1, partials);
    reduce_k<<<1, 128, 0, stream>>>(partials, stats, NSTAT);
    bnprep_k<<<1, 64, 0, stream>>>(stats, g1, be1, bnp);
    // 3) BN1 + ReLU in place
    bnrelu_k<<<1024, 256, 0, stream>>>(buf1, bnp);
    // 4) offset conv3x3 -> offsbuf
    offsetconv_k<<<NPIX / 128, 128, 0, stream>>>(buf1, ow, ob, offsbuf);
    // 5) deformable conv3x3 (bilinear gather + WMMA GEMM) -> buf2
    deform_k<<<NTILE, 128, 0, stream>>>(buf1, offsbuf, dw, db, buf2);
    // 6) conv1x1 #2 (WMMA) -> buf1 (y2, NHWC)
    conv2_k<<<NTILE, 128, 0, stream>>>(buf2, w2, b2, buf1);
    // 7) BN2 stats + prep
    stats_k<<<NSTAT, 256, 0, stream>>>(buf1, partials);
    reduce_k<<<1, 128, 0, stream>>>(partials, stats + 128, NSTAT);
    bnprep_k<<<1, 64, 0, stream>>>(stats + 128, g2, be2, bnp + 128);
    // 8) BN2 apply + NHWC->NCHW transpose -> d_out
    final_k<<<NTILE, 256, 0, stream>>>(buf1, bnp + 128, out);
}